// CompositionNetwork_39041252720734
// MI455X (gfx1250) — compile-verified
//
#include <hip/hip_runtime.h>

// Embedding-bag sum pooling (masked gather + sum) for MI455X / gfx1250.
//   target     : [B, HIST] int32 (sentinel -1 = dropped)
//   emb_weight : [N_EMB, REPR] f32
//   out        : [B, 1, REPR] f32
//
// Memory-bound gather (~670MB reads / 16MB writes; HBM 23.3 TB/s => ~30us floor,
// less with L2 reuse: 205MB table vs 192MB L2, each row touched ~3.3x).
// Strategy: one workgroup per batch row; uniform indices loaded as int2 pairs
// (s_load_b64, halves KMcnt stalls); each lane async-copies its 16B slice of
// each 2KB embedding row into an 8-deep LDS ring via
// global_load_async_to_lds_b128 (ASYNCcnt-tracked, no dest VGPRs, no barriers),
// accumulates from LDS (ds_load_b128), and stores the result non-temporally so
// the write stream does not evict embedding rows from L2.

#define REPR    512
#define HIST    50
#define STAGES  8              // async pipeline depth (16KB LDS per block)
#define THREADS 128            // 4 waves * 32 lanes; lane covers 16B of a row

typedef float v4f __attribute__((ext_vector_type(4)));

#define WAIT_ASYNCCNT(n) asm volatile("s_wait_asynccnt " #n ::: "memory")

// Flat pointer to LDS: AS3->flat lowering places the wave-relative LDS byte
// offset in the low 32 bits (ISA 10.2).
__device__ __forceinline__ unsigned lds_off(const void* p) {
  return (unsigned)(unsigned long long)p;
}

// GVS-mode async copy: copies 16 bytes / lane from global address saddr plus offset voff to the LDS address in lds_addr.
__device__ __forceinline__ void async_b128_to_lds(unsigned lds_addr,
                                                  unsigned voff,
                                                  const float* __restrict__ saddr) {
  asm volatile("global_load_async_to_lds_b128 %0, %1, %2"
               :
               : "v"(lds_addr), "v"(voff), "s"(saddr)
               : "memory");
}

__global__ __launch_bounds__(THREADS, 1)
void emb_pool_async_kernel(const int* __restrict__ target,
                           const float* __restrict__ emb,
                           float* __restrict__ out,
                           int hist) {
  __shared__ __align__(16) float buf[STAGES * REPR];

  const int b    = blockIdx.x;
  const int tid  = threadIdx.x;                      // 0..127
  const int elem = tid << 2;                         // first float owned by lane
  const unsigned lane_bytes = (unsigned)elem * 4u;   // 16B per lane

  const int* __restrict__ trow = target + b * hist;  // uniform

  v4f acc = (v4f)0.0f;

  int wr = 0, rd = 0;

  // Process one gathered index: issue async copy; consume oldest when ring full.
#define PROCESS_IDX(vidx)                                                     \
  do {                                                                        \
    const int v_ = (vidx);                                                    \
    if (v_ >= 0) {                                                            \
      const unsigned slot_ = (unsigned)wr & (STAGES - 1);                     \
      const unsigned la_   = lds_off(&buf[slot_ * REPR + elem]);              \
      const unsigned go_   = (unsigned)v_ * (REPR * 4u) + lane_bytes;         \
      async_b128_to_lds(la_, go_, emb);                                       \
      ++wr;                                                                   \
      if (wr - rd == STAGES) {                                                \
        WAIT_ASYNCCNT(7); /* oldest of 8 complete (in-order) */               \
        const unsigned rs_ = (unsigned)rd & (STAGES - 1);                     \
        const v4f vv_ =                                                       \
            *reinterpret_cast<const v4f*>(&buf[rs_ * REPR + elem]);           \
        acc += vv_;                                                           \
        ++rd;                                                                 \
      }                                                                       \
    }                                                                         \
  } while (0)

  // Uniform indices fetched two at a time: rows start at 200*b bytes (8B
  // aligned), so int2 loads are safe -> s_load_b64, half the KMcnt stalls.
  const int pairs = hist >> 1;
#pragma unroll 1
  for (int p = 0; p < pairs; ++p) {
    const int2 iv = *reinterpret_cast<const int2*>(trow + 2 * p);
    PROCESS_IDX(iv.x);
    PROCESS_IDX(iv.y);
  }
  if (hist & 1) {
    PROCESS_IDX(trow[hist - 1]);
  }
#undef PROCESS_IDX

  // Drain: everything outstanding lands, then consume remaining slots.
  WAIT_ASYNCCNT(0);
#pragma unroll 1
  for (; rd < wr; ++rd) {
    const unsigned rs = (unsigned)rd & (STAGES - 1);
    acc += *reinterpret_cast<const v4f*>(&buf[rs * REPR + elem]);
  }

  // Non-temporal store: output is write-once, keep L2 for the embedding table.
  __builtin_nontemporal_store(acc,
      reinterpret_cast<v4f*>(out + (size_t)b * REPR + elem));
}

extern "C" void kernel_launch(void* const* d_in, const int* in_sizes, int n_in,
                              void* d_out, int out_size, void* d_ws, size_t ws_size,
                              hipStream_t stream) {
  (void)n_in; (void)d_ws; (void)ws_size; (void)out_size;
  const int* target = (const int*)d_in[0];      // [B, HIST] int
  const float* emb  = (const float*)d_in[1];    // [N_EMB, REPR] f32
  float* out        = (float*)d_out;            // [B, 1, REPR] f32

  const int batch = in_sizes[0] / HIST;         // 8192

  dim3 grid(batch);
  dim3 block(THREADS);
  emb_pool_async_kernel<<<grid, block, 0, stream>>>(target, emb, out, HIST);
}